// DecoderRNN_2310692405976
// MI455X (gfx1250) — compile-verified
//
#include <hip/hip_runtime.h>
#include <math.h>

// Problem constants (fixed by the reference)
#define BB   64
#define EE   512
#define HH   512
#define SS   400
#define TT   50
#define VV   50000
#define K3H  1536
#define FEPS 1e-8f

typedef float v2f __attribute__((ext_vector_type(2)));
typedef float v8f __attribute__((ext_vector_type(8)));

__device__ __forceinline__ float waveReduceSum(float v) {
  #pragma unroll
  for (int off = 16; off > 0; off >>= 1)
    v += __shfl_down(v, off, 32);
  return v;
}

// ---------------- 1. GRU single step ----------------
__global__ void __launch_bounds__(256)
gru_kernel(const float* __restrict__ emb, const float* __restrict__ hid,
           const float* __restrict__ wih, const float* __restrict__ whh,
           const float* __restrict__ bih, const float* __restrict__ bhh,
           float* __restrict__ hnew, float* __restrict__ hnew_out) {
  int t = blockIdx.x * blockDim.x + threadIdx.x;
  if (t >= BB * HH) return;
  int b = t / HH, j = t % HH;
  const float* x = emb + (size_t)b * EE;
  const float* h = hid + (size_t)b * HH;
  float gi_r = bih[j], gi_z = bih[HH + j], gi_n = bih[2 * HH + j];
  float gh_r = bhh[j], gh_z = bhh[HH + j], gh_n = bhh[2 * HH + j];
  const float* wr = wih + (size_t)j * EE;
  const float* wz = wih + (size_t)(HH + j) * EE;
  const float* wn = wih + (size_t)(2 * HH + j) * EE;
  #pragma unroll 4
  for (int k = 0; k < EE; k++) {
    float xv = x[k];
    gi_r += wr[k] * xv; gi_z += wz[k] * xv; gi_n += wn[k] * xv;
  }
  const float* ur = whh + (size_t)j * HH;
  const float* uz = whh + (size_t)(HH + j) * HH;
  const float* un = whh + (size_t)(2 * HH + j) * HH;
  #pragma unroll 4
  for (int k = 0; k < HH; k++) {
    float hv = h[k];
    gh_r += ur[k] * hv; gh_z += uz[k] * hv; gh_n += un[k] * hv;
  }
  float r = 1.f / (1.f + expf(-(gi_r + gh_r)));
  float z = 1.f / (1.f + expf(-(gi_z + gh_z)));
  float n = tanhf(gi_n + r * gh_n);
  float hn = (1.f - z) * n + z * h[j];
  hnew[t] = hn;
  hnew_out[t] = hn;
}

// ---------------- 2. bilinear projections hw = h_new @ W ----------------
__global__ void __launch_bounds__(256)
hw_kernel(const float* __restrict__ hnew, const float* __restrict__ wenc,
          const float* __restrict__ wdec, float* __restrict__ hwe,
          float* __restrict__ hwd) {
  int t = blockIdx.x * blockDim.x + threadIdx.x;
  if (t >= BB * HH) return;
  int b = t / HH, j = t % HH;
  const float* h = hnew + (size_t)b * HH;
  float se = 0.f, sd = 0.f;
  #pragma unroll 4
  for (int k = 0; k < HH; k++) {
    float hv = h[k];
    se += wenc[(size_t)k * HH + j] * hv;
    sd += wdec[(size_t)k * HH + j] * hv;
  }
  hwe[t] = se;
  hwd[t] = sd;
}

// ---------------- 3. attention energies (one wave per (s,b)) ----------------
__global__ void __launch_bounds__(256)
energy_kernel(const float* __restrict__ hw, const float* __restrict__ states,
              const float* __restrict__ biasPtr, const float* __restrict__ cov,
              const float* __restrict__ cwPtr, float* __restrict__ energy, int Sdim) {
  int wave = (blockIdx.x * blockDim.x + threadIdx.x) >> 5;
  int lane = threadIdx.x & 31;
  if (wave >= Sdim * BB) return;
  int s = wave / BB, b = wave % BB;
  const float* st = states + ((size_t)s * BB + b) * HH;
  const float* hv = hw + (size_t)b * HH;
  float acc = 0.f;
  for (int k = lane; k < HH; k += 32) acc += hv[k] * st[k];
  acc = waveReduceSum(acc);
  if (lane == 0) {
    float e = acc + biasPtr[0];
    if (cov) e += cwPtr[0] * logf(cov[(size_t)b * SS + s] + FEPS);
    energy[(size_t)s * BB + b] = e;
  }
}

// ---------------- 4. softmax over dim 0, per column b ----------------
__global__ void __launch_bounds__(256)
softmax_col_kernel(float* __restrict__ energy, int Sdim, float* __restrict__ attnT) {
  int b = blockIdx.x;
  __shared__ float red[256];
  float m = -INFINITY;
  for (int s = threadIdx.x; s < Sdim; s += blockDim.x)
    m = fmaxf(m, energy[(size_t)s * BB + b]);
  red[threadIdx.x] = m; __syncthreads();
  for (int o = blockDim.x / 2; o > 0; o >>= 1) {
    if (threadIdx.x < o) red[threadIdx.x] = fmaxf(red[threadIdx.x], red[threadIdx.x + o]);
    __syncthreads();
  }
  m = red[0]; __syncthreads();
  float sum = 0.f;
  for (int s = threadIdx.x; s < Sdim; s += blockDim.x) {
    float e = expf(energy[(size_t)s * BB + b] - m);
    energy[(size_t)s * BB + b] = e;
    sum += e;
  }
  red[threadIdx.x] = sum; __syncthreads();
  for (int o = blockDim.x / 2; o > 0; o >>= 1) {
    if (threadIdx.x < o) red[threadIdx.x] += red[threadIdx.x + o];
    __syncthreads();
  }
  float inv = 1.f / red[0];
  for (int s = threadIdx.x; s < Sdim; s += blockDim.x) {
    float a = energy[(size_t)s * BB + b] * inv;
    energy[(size_t)s * BB + b] = a;
    if (attnT) attnT[(size_t)b * Sdim + s] = a;
  }
}

// ---------------- 5. weighted context ----------------
__global__ void __launch_bounds__(256)
context_kernel(const float* __restrict__ states, const float* __restrict__ attn,
               float* __restrict__ ctx, int Sdim) {
  int t = blockIdx.x * blockDim.x + threadIdx.x;
  if (t >= BB * HH) return;
  int b = t / HH, j = t % HH;
  float acc = 0.f;
  for (int s = 0; s < Sdim; s++)
    acc += states[((size_t)s * BB + b) * HH + j] * attn[(size_t)s * BB + b];
  ctx[t] = acc;
}

// ---------------- 6. concat [h_new, enc_ctx, dec_ctx] ----------------
__global__ void __launch_bounds__(256)
combine_kernel(const float* __restrict__ hnew, const float* __restrict__ ectx,
               const float* __restrict__ dctx, float* __restrict__ comb) {
  int t = blockIdx.x * blockDim.x + threadIdx.x;
  if (t >= BB * K3H) return;
  int b = t / K3H, j = t % K3H;
  float v;
  if (j < HH)           v = hnew[(size_t)b * HH + j];
  else if (j < 2 * HH)  v = ectx[(size_t)b * HH + j - HH];
  else                  v = dctx[(size_t)b * HH + j - 2 * HH];
  comb[t] = v;
}

// ---------------- 7. pointer gate ----------------
__global__ void __launch_bounds__(32)
ptr_kernel(const float* __restrict__ comb, const float* __restrict__ pw,
           const float* __restrict__ pb, float* __restrict__ prob_ptr,
           float* __restrict__ out_ptr) {
  int b = blockIdx.x, lane = threadIdx.x;
  float acc = 0.f;
  for (int k = lane; k < K3H; k += 32) acc += comb[(size_t)b * K3H + k] * pw[k];
  acc = waveReduceSum(acc);
  if (lane == 0) {
    float p = 1.f / (1.f + expf(-(acc + pb[0])));
    prob_ptr[b] = p;
    out_ptr[b] = p;
  }
}

// ---------------- 8. logits GEMM via V_WMMA_F32_16X16X4_F32 ----------------
// One wave per 16-wide strip of V; each wave accumulates all 4 M-tiles (B=64)
// so each out_w fragment load is reused by 4 WMMAs. A(16x4) and B(4x16) f32
// fragments: lane l<16 holds {K0,K1} of row l; lane>=16 holds {K2,K3}.
__global__ void __launch_bounds__(128)
logits_wmma_kernel(const float* __restrict__ comb, const float* __restrict__ outw,
                   const float* __restrict__ outb, float* __restrict__ logits) {
  int waveInBlock = threadIdx.x >> 5;
  int lane = threadIdx.x & 31;
  int nt = blockIdx.x * 4 + waveInBlock;
  if (nt >= VV / 16) return;          // wave-uniform: EXEC stays all-ones
  int v0 = nt * 16;
  int half = (lane >> 4) & 1;         // 0 -> K0..1, 1 -> K2..3
  int l15 = lane & 15;

  const float* bp  = outw + (size_t)(v0 + l15) * K3H + half * 2;
  const float* ap0 = comb + (size_t)(0  + l15) * K3H + half * 2;
  const float* ap1 = comb + (size_t)(16 + l15) * K3H + half * 2;
  const float* ap2 = comb + (size_t)(32 + l15) * K3H + half * 2;
  const float* ap3 = comb + (size_t)(48 + l15) * K3H + half * 2;

  v8f acc0 = {}, acc1 = {}, acc2 = {}, acc3 = {};
  #pragma unroll 4
  for (int k = 0; k < K3H; k += 4) {
    v2f bf = *(const v2f*)(bp + k);
    v2f a0 = *(const v2f*)(ap0 + k);
    v2f a1 = *(const v2f*)(ap1 + k);
    v2f a2 = *(const v2f*)(ap2 + k);
    v2f a3 = *(const v2f*)(ap3 + k);
    acc0 = __builtin_amdgcn_wmma_f32_16x16x4_f32(false, a0, false, bf, (short)0, acc0, false, false);
    acc1 = __builtin_amdgcn_wmma_f32_16x16x4_f32(false, a1, false, bf, (short)0, acc1, false, false);
    acc2 = __builtin_amdgcn_wmma_f32_16x16x4_f32(false, a2, false, bf, (short)0, acc2, false, false);
    acc3 = __builtin_amdgcn_wmma_f32_16x16x4_f32(false, a3, false, bf, (short)0, acc3, false, false);
  }
  // D layout: VGPR i, lanes 0-15 -> (M=i, N=lane); lanes 16-31 -> (M=i+8, N=lane-16)
  int col = l15;
  float bias = outb[v0 + col];
  #pragma unroll
  for (int i = 0; i < 8; i++) {
    int rowl = i + half * 8;
    logits[(size_t)(0  + rowl) * VV + v0 + col] = acc0[i] + bias;
    logits[(size_t)(16 + rowl) * VV + v0 + col] = acc1[i] + bias;
    logits[(size_t)(32 + rowl) * VV + v0 + col] = acc2[i] + bias;
    logits[(size_t)(48 + rowl) * VV + v0 + col] = acc3[i] + bias;
  }
}

// ---------------- 9. per-row max & sum(exp) over V ----------------
__global__ void __launch_bounds__(256)
rowstat_kernel(const float* __restrict__ logits, float* __restrict__ rowmax,
               float* __restrict__ rowinv) {
  int b = blockIdx.x;
  __shared__ float red[256];
  float m = -INFINITY;
  for (int v = threadIdx.x; v < VV; v += blockDim.x)
    m = fmaxf(m, logits[(size_t)b * VV + v]);
  red[threadIdx.x] = m; __syncthreads();
  for (int o = blockDim.x / 2; o > 0; o >>= 1) {
    if (threadIdx.x < o) red[threadIdx.x] = fmaxf(red[threadIdx.x], red[threadIdx.x + o]);
    __syncthreads();
  }
  m = red[0]; __syncthreads();
  float sum = 0.f;
  for (int v = threadIdx.x; v < VV; v += blockDim.x)
    sum += expf(logits[(size_t)b * VV + v] - m);
  red[threadIdx.x] = sum; __syncthreads();
  for (int o = blockDim.x / 2; o > 0; o >>= 1) {
    if (threadIdx.x < o) red[threadIdx.x] += red[threadIdx.x + o];
    __syncthreads();
  }
  if (threadIdx.x == 0) { rowmax[b] = m; rowinv[b] = 1.f / red[0]; }
}

// ---------------- 10. output[:, :V] = (1-p)*softmax, rest zero ----------------
__global__ void __launch_bounds__(256)
writeout_kernel(const float* __restrict__ logits, const float* __restrict__ rowmax,
                const float* __restrict__ rowinv, const float* __restrict__ prob_ptr,
                float* __restrict__ out, int ext) {
  size_t t = (size_t)blockIdx.x * blockDim.x + threadIdx.x;
  if (t >= (size_t)BB * ext) return;
  int b = (int)(t / ext);
  int v = (int)(t % ext);
  float val = 0.f;
  if (v < VV)
    val = (1.f - prob_ptr[b]) * expf(logits[(size_t)b * VV + v] - rowmax[b]) * rowinv[b];
  out[t] = val;
}

// ---------------- 11. scatter-add pointer mass ----------------
__global__ void __launch_bounds__(256)
scatter_kernel(const int* __restrict__ idx, const float* __restrict__ attn,
               const float* __restrict__ prob_ptr, float* __restrict__ out, int ext) {
  int t = blockIdx.x * blockDim.x + threadIdx.x;
  if (t >= SS * BB) return;
  int s = t / BB, b = t % BB;
  int id = idx[(size_t)s * BB + b];
  atomicAdd(out + (size_t)b * ext + id, prob_ptr[b] * attn[(size_t)s * BB + b]);
}

// ---------------- 12. final log ----------------
__global__ void __launch_bounds__(256)
log_kernel(float* __restrict__ out, int ext) {
  size_t t = (size_t)blockIdx.x * blockDim.x + threadIdx.x;
  if (t < (size_t)BB * ext) out[t] = logf(out[t] + FEPS);
}

extern "C" void kernel_launch(void* const* d_in, const int* in_sizes, int n_in,
                              void* d_out, int out_size, void* d_ws, size_t ws_size,
                              hipStream_t stream) {
  const float* emb   = (const float*)d_in[0];
  const float* hid   = (const float*)d_in[1];
  const float* encS  = (const float*)d_in[2];
  const float* decS  = (const float*)d_in[3];
  const float* cov   = (const float*)d_in[4];
  const int*   widx  = (const int*)  d_in[5];
  const float* wih   = (const float*)d_in[6];
  const float* whh   = (const float*)d_in[7];
  const float* bih   = (const float*)d_in[8];
  const float* bhh   = (const float*)d_in[9];
  const float* encW  = (const float*)d_in[10];
  const float* encB  = (const float*)d_in[11];
  const float* cw    = (const float*)d_in[12];
  const float* decW  = (const float*)d_in[13];
  const float* decB  = (const float*)d_in[14];
  const float* ptrW  = (const float*)d_in[15];
  const float* ptrB  = (const float*)d_in[16];
  const float* outW  = (const float*)d_in[17];
  const float* outB  = (const float*)d_in[18];

  const int ext = (out_size - BB * HH - BB * SS - BB) / BB;   // 50100

  float* outp = (float*)d_out;
  float* out_hnew = outp + (size_t)BB * ext;
  float* out_attn = out_hnew + (size_t)BB * HH;
  float* out_pptr = out_attn + (size_t)BB * SS;

  // workspace layout (floats)
  float* wf = (float*)d_ws;
  float* hnew    = wf;                       // 32768
  float* hwe     = hnew    + BB * HH;        // 32768
  float* hwd     = hwe     + BB * HH;        // 32768
  float* eattn   = hwd     + BB * HH;        // 25600 (S-major)
  float* dattn   = eattn   + SS * BB;        // 3200
  float* ectx    = dattn   + TT * BB;        // 32768
  float* dctx    = ectx    + BB * HH;        // 32768
  float* comb    = dctx    + BB * HH;        // 98304
  float* pptr    = comb    + BB * K3H;       // 64
  float* rowmax  = pptr    + BB;             // 64
  float* rowinv  = rowmax  + BB;             // 64
  float* logits  = rowinv  + BB;             // 64*50000

  hipLaunchKernelGGL(gru_kernel, dim3((BB * HH + 255) / 256), dim3(256), 0, stream,
                     emb, hid, wih, whh, bih, bhh, hnew, out_hnew);
  hipLaunchKernelGGL(hw_kernel, dim3((BB * HH + 255) / 256), dim3(256), 0, stream,
                     hnew, encW, decW, hwe, hwd);
  hipLaunchKernelGGL(energy_kernel, dim3((SS * BB * 32 + 255) / 256), dim3(256), 0, stream,
                     hwe, encS, encB, cov, cw, eattn, SS);
  hipLaunchKernelGGL(energy_kernel, dim3((TT * BB * 32 + 255) / 256), dim3(256), 0, stream,
                     hwd, decS, decB, (const float*)nullptr, cw, dattn, TT);
  hipLaunchKernelGGL(softmax_col_kernel, dim3(BB), dim3(256), 0, stream, eattn, SS, out_attn);
  hipLaunchKernelGGL(softmax_col_kernel, dim3(BB), dim3(256), 0, stream, dattn, TT, (float*)nullptr);
  hipLaunchKernelGGL(context_kernel, dim3((BB * HH + 255) / 256), dim3(256), 0, stream,
                     encS, eattn, ectx, SS);
  hipLaunchKernelGGL(context_kernel, dim3((BB * HH + 255) / 256), dim3(256), 0, stream,
                     decS, dattn, dctx, TT);
  hipLaunchKernelGGL(combine_kernel, dim3((BB * K3H + 255) / 256), dim3(256), 0, stream,
                     hnew, ectx, dctx, comb);
  hipLaunchKernelGGL(ptr_kernel, dim3(BB), dim3(32), 0, stream, comb, ptrW, ptrB, pptr, out_pptr);
  hipLaunchKernelGGL(logits_wmma_kernel, dim3((VV / 16 + 3) / 4), dim3(128), 0, stream,
                     comb, outW, outB, logits);
  hipLaunchKernelGGL(rowstat_kernel, dim3(BB), dim3(256), 0, stream, logits, rowmax, rowinv);
  size_t nOut = (size_t)BB * ext;
  hipLaunchKernelGGL(writeout_kernel, dim3((unsigned)((nOut + 255) / 256)), dim3(256), 0, stream,
                     logits, rowmax, rowinv, pptr, outp, ext);
  hipLaunchKernelGGL(scatter_kernel, dim3((SS * BB + 255) / 256), dim3(256), 0, stream,
                     widx, eattn, pptr, outp, ext);
  hipLaunchKernelGGL(log_kernel, dim3((unsigned)((nOut + 255) / 256)), dim3(256), 0, stream,
                     outp, ext);
}